// SlotAttention_28484223107765
// MI455X (gfx1250) — compile-verified
//
#include <hip/hip_runtime.h>
#include <hip/hip_bf16.h>

typedef __attribute__((ext_vector_type(2))) float v2f;
typedef __attribute__((ext_vector_type(8))) float v8f;

#define EPSF 1e-8f

// ---------------------------------------------------------------------------
// Kernel 1: LayerNorm(inputs) + K/V projection via f32 WMMA + wi logits
// Block: 256 threads (8 waves), 128 rows of the flattened [B*N, 64] input.
// ---------------------------------------------------------------------------
__global__ __launch_bounds__(256) void k_ln_proj(
    const float* __restrict__ inp,
    const float* __restrict__ Wk, const float* __restrict__ Wv,
    const float* __restrict__ wi_w, const float* __restrict__ wi_b,
    const float* __restrict__ ln_g, const float* __restrict__ ln_b,
    float* __restrict__ kout, float* __restrict__ vout,
    float* __restrict__ logit)
{
  __shared__ float xln[128][68];
  const int rowbase = blockIdx.x * 128;
  const int tid = threadIdx.x;
  if (tid < 128) {
    const float* r = inp + (size_t)(rowbase + tid) * 64;
    float m = 0.f;
#pragma unroll
    for (int i = 0; i < 64; ++i) m += r[i];
    m *= (1.f / 64.f);
    float var = 0.f;
#pragma unroll
    for (int i = 0; i < 64; ++i) { float d = r[i] - m; var += d * d; }
    var *= (1.f / 64.f);
    const float rs = rsqrtf(var + 1e-5f);
    float lg = 0.f;
#pragma unroll
    for (int i = 0; i < 64; ++i) {
      const float y = (r[i] - m) * rs * ln_g[i] + ln_b[i];
      xln[tid][i] = y;
      lg += y * wi_w[i];
    }
    logit[rowbase + tid] = lg + wi_b[0];
  }
  __syncthreads();

  const int lane = tid & 31;
  const int wave = tid >> 5;
  const int l = lane & 15, g = lane >> 4;
  const int m0 = wave * 16;
  for (int nt = 0; nt < 4; ++nt) {
    const int n0 = nt * 16;
    v8f ck = {}; v8f cv = {};
    for (int k0 = 0; k0 < 64; k0 += 4) {
      v2f a;  a.x  = xln[m0 + l][k0 + 2 * g];       a.y  = xln[m0 + l][k0 + 2 * g + 1];
      v2f bk; bk.x = Wk[(n0 + l) * 64 + k0 + 2 * g]; bk.y = Wk[(n0 + l) * 64 + k0 + 2 * g + 1];
      v2f bv; bv.x = Wv[(n0 + l) * 64 + k0 + 2 * g]; bv.y = Wv[(n0 + l) * 64 + k0 + 2 * g + 1];
      ck = __builtin_amdgcn_wmma_f32_16x16x4_f32(false, a, false, bk, (short)0, ck, false, false);
      cv = __builtin_amdgcn_wmma_f32_16x16x4_f32(false, a, false, bv, (short)0, cv, false, false);
    }
#pragma unroll
    for (int i = 0; i < 8; ++i) {
      const size_t row = (size_t)rowbase + m0 + i + 8 * g;
      kout[row * 64 + n0 + l] = ck[i];
      vout[row * 64 + n0 + l] = cv[i];
    }
  }
}

// ---------------------------------------------------------------------------
// Kernel 2: a = softmax(logits over N)*NS ; store log(a + eps)
// ---------------------------------------------------------------------------
__global__ __launch_bounds__(256) void k_log_a(const float* __restrict__ logit,
                                               float* __restrict__ log_a)
{
  __shared__ float red[256];
  const int b = blockIdx.x, t = threadIdx.x;
  const float* L = logit + b * 1024;
  float mx = -1e30f;
  for (int i = t; i < 1024; i += 256) mx = fmaxf(mx, L[i]);
  red[t] = mx; __syncthreads();
  for (int s = 128; s > 0; s >>= 1) { if (t < s) red[t] = fmaxf(red[t], red[t + s]); __syncthreads(); }
  mx = red[0]; __syncthreads();
  float sm = 0.f;
  for (int i = t; i < 1024; i += 256) sm += __expf(L[i] - mx);
  red[t] = sm; __syncthreads();
  for (int s = 128; s > 0; s >>= 1) { if (t < s) red[t] += red[t + s]; __syncthreads(); }
  sm = red[0];
  for (int i = t; i < 1024; i += 256) {
    const float a = __expf(L[i] - mx) / sm * 8.0f;
    log_a[b * 1024 + i] = __logf(a + EPSF);
  }
}

// ---------------------------------------------------------------------------
// Kernel 3: slots = mu + (|sigma|+eps)*noise
// ---------------------------------------------------------------------------
__global__ __launch_bounds__(256) void k_slots_init(
    const float* __restrict__ mu, const float* __restrict__ sigma,
    const float* __restrict__ noise, float* __restrict__ slots)
{
  const int i = blockIdx.x * 256 + threadIdx.x;
  const int d = i & 63;
  slots[i] = mu[d] + (fabsf(sigma[d]) + EPSF) * noise[i];
}

// ---------------------------------------------------------------------------
// log-domain Sinkhorn forward (8 iters) for one batch, plan [1024 x 8] in LDS.
// Optionally records u_t / v_t history for the analytic backward pass.
// ---------------------------------------------------------------------------
__device__ __forceinline__ void block_sink_fwd(
    float* __restrict__ Cs, const float* __restrict__ la, const float* __restrict__ lb,
    float* __restrict__ u, float* __restrict__ vv,
    float* __restrict__ uh, float* __restrict__ vh,
    float* __restrict__ red, int tid, bool warm)
{
  if (!warm && tid < 8) vv[tid] = 0.f;
  __syncthreads();
  if (uh != nullptr && tid < 8) vh[tid] = vv[tid];
  __syncthreads();
  for (int t = 1; t <= 8; ++t) {
    // u = log_a - LSE_s(v - C)
    for (int j = 0; j < 4; ++j) {
      const int n = tid + j * 256;
      float mx = -1e30f;
#pragma unroll
      for (int s = 0; s < 8; ++s) mx = fmaxf(mx, vv[s] - Cs[n * 8 + s]);
      float sm = 0.f;
#pragma unroll
      for (int s = 0; s < 8; ++s) sm += __expf(vv[s] - Cs[n * 8 + s] - mx);
      const float un = la[n] - (mx + __logf(sm));
      u[n] = un;
      if (uh != nullptr) uh[(t - 1) * 1024 + n] = un;
    }
    __syncthreads();
    // v = log_b - LSE_n(u - C): two-pass (max, sum-exp) block reduction
    float pm[8];
#pragma unroll
    for (int s = 0; s < 8; ++s) pm[s] = -1e30f;
    for (int j = 0; j < 4; ++j) {
      const int n = tid + j * 256;
      const float un = u[n];
#pragma unroll
      for (int s = 0; s < 8; ++s) pm[s] = fmaxf(pm[s], un - Cs[n * 8 + s]);
    }
#pragma unroll
    for (int s = 0; s < 8; ++s) red[tid * 8 + s] = pm[s];
    __syncthreads();
    for (int st = 128; st > 0; st >>= 1) {
      if (tid < st)
#pragma unroll
        for (int s = 0; s < 8; ++s)
          red[tid * 8 + s] = fmaxf(red[tid * 8 + s], red[(tid + st) * 8 + s]);
      __syncthreads();
    }
    float ms[8];
#pragma unroll
    for (int s = 0; s < 8; ++s) ms[s] = red[s];
    __syncthreads();
    float ps[8];
#pragma unroll
    for (int s = 0; s < 8; ++s) ps[s] = 0.f;
    for (int j = 0; j < 4; ++j) {
      const int n = tid + j * 256;
      const float un = u[n];
#pragma unroll
      for (int s = 0; s < 8; ++s) ps[s] += __expf(un - Cs[n * 8 + s] - ms[s]);
    }
#pragma unroll
    for (int s = 0; s < 8; ++s) red[tid * 8 + s] = ps[s];
    __syncthreads();
    for (int st = 128; st > 0; st >>= 1) {
      if (tid < st)
#pragma unroll
        for (int s = 0; s < 8; ++s) red[tid * 8 + s] += red[(tid + st) * 8 + s];
      __syncthreads();
    }
    if (tid < 8) {
      vv[tid] = lb[tid] - (ms[tid] + __logf(red[tid]));
      if (uh != nullptr) vh[t * 8 + tid] = vv[tid];
    }
    __syncthreads();
  }
}

// ---------------------------------------------------------------------------
// Kernel 4: one workgroup per batch — q projection, cdist (WMMA), MESH
// (forward + analytic reverse-mode Sinkhorn), final warm Sinkhorn, then
// updates = P^T v via async global->LDS staged tiles, GRU, MLP.
// Dynamic LDS: ~164 KB (fits the 320 KB/WGP budget).
// ---------------------------------------------------------------------------
__global__ __launch_bounds__(256) void k_slot_iter(
    const float* __restrict__ kG, const float* __restrict__ vG,
    const float* __restrict__ log_aG, float* __restrict__ slotsG,
    const float* __restrict__ Wq,
    const float* __restrict__ ws_w, const float* __restrict__ ws_b,
    const float* __restrict__ ln_s_g, const float* __restrict__ ln_s_b,
    const float* __restrict__ w_ih, const float* __restrict__ w_hh,
    const float* __restrict__ b_ih, const float* __restrict__ b_hh,
    const float* __restrict__ fc1_w, const float* __restrict__ fc1_b,
    const float* __restrict__ fc2_w, const float* __restrict__ fc2_b,
    const float* __restrict__ ln_f_g, const float* __restrict__ ln_f_b,
    float* __restrict__ out_slots, float* __restrict__ out_pos,
    float* __restrict__ out_attn, int is_last)
{
  extern __shared__ float smem[];
  float* Cs    = smem;            // 8192  cost matrix [1024][8]
  float* gC    = Cs + 8192;       // 8192  dH/dC
  float* Pm    = gC + 8192;       // 8192  transport plan
  float* uh    = Pm + 8192;       // 8192  u_t history; reused as v-tile later
  float* u     = uh + 8192;       // 1024
  float* la    = u + 1024;        // 1024  log_a
  float* kk    = la + 1024;       // 1024  |k_n|^2
  float* red   = kk + 1024;       // 2048  reduction scratch [256][8]
  float* qs    = red + 2048;      // 1024  q padded [16][64]; reused as ff[8][128]
  float* sln   = qs + 1024;       // 512   layernormed slots
  float* sprev = sln + 512;       // 512
  float* upd   = sprev + 512;     // 512
  float* hnew  = upd + 512;       // 512
  float* vh    = hnew + 512;      // 80    v_t history (t=0..8)
  float* vv    = vh + 80;         // 8
  float* lb    = vv + 8;          // 8     log_b
  float* gvb   = lb + 8;          // 8
  float* misc  = gvb + 8;         // 64

  const int b = blockIdx.x;
  const int tid = threadIdx.x;
  const size_t kvbase = (size_t)b * 1024 * 64;

  // --- Phase A: log_a and |k|^2 per row ---
  for (int j = 0; j < 4; ++j) {
    const int n = tid + j * 256;
    la[n] = log_aG[b * 1024 + n];
    const float* kr = kG + kvbase + (size_t)n * 64;
    float s2 = 0.f;
#pragma unroll
    for (int d = 0; d < 64; ++d) s2 += kr[d] * kr[d];
    kk[n] = s2;
  }
  // --- Phase B: slots, layernorm, bm softmax, q projection ---
  for (int j = 0; j < 2; ++j) {
    const int idx = tid + j * 256;
    sprev[idx] = slotsG[(size_t)b * 512 + idx];
  }
  __syncthreads();
  if (tid < 8) {
    float m = 0.f;
#pragma unroll
    for (int i = 0; i < 64; ++i) m += sprev[tid * 64 + i];
    m *= (1.f / 64.f);
    float var = 0.f;
#pragma unroll
    for (int i = 0; i < 64; ++i) { const float d = sprev[tid * 64 + i] - m; var += d * d; }
    var *= (1.f / 64.f);
    const float rs = rsqrtf(var + 1e-5f);
    float lg = 0.f;
#pragma unroll
    for (int i = 0; i < 64; ++i) {
      const float y = (sprev[tid * 64 + i] - m) * rs * ln_s_g[i] + ln_s_b[i];
      sln[tid * 64 + i] = y;
      lg += y * ws_w[i];
    }
    misc[tid] = lg + ws_b[0];
  }
  __syncthreads();
  if (tid == 0) {
    float mx = -1e30f;
    for (int s = 0; s < 8; ++s) mx = fmaxf(mx, misc[s]);
    float sm = 0.f;
    for (int s = 0; s < 8; ++s) sm += __expf(misc[s] - mx);
    for (int s = 0; s < 8; ++s) {
      const float bmv = __expf(misc[s] - mx) / sm * 8.0f;
      lb[s] = __logf(bmv + EPSF);
    }
  }
  // q = sln @ Wq^T, padded to 16 slot rows (rows 8..15 zero)
  for (int j = 0; j < 4; ++j) {
    const int idx = tid + j * 256;
    const int s = idx >> 6, d = idx & 63;
    if (s < 8) {
      float acc = 0.f;
#pragma unroll
      for (int e = 0; e < 64; ++e) acc += sln[s * 64 + e] * Wq[d * 64 + e];
      qs[idx] = acc;
    } else {
      qs[idx] = 0.f;
    }
  }
  __syncthreads();
  if (tid < 8) {
    float s2 = 0.f;
#pragma unroll
    for (int d = 0; d < 64; ++d) s2 += qs[tid * 64 + d] * qs[tid * 64 + d];
    misc[8 + tid] = s2;
  }
  __syncthreads();

  // --- Phase C: C = cdist(k, q) via WMMA (k.q dot products) ---
  {
    const int lane = tid & 31, wave = tid >> 5;
    const int l = lane & 15, g = lane >> 4;
    for (int jt = 0; jt < 8; ++jt) {
      const int m0 = (wave * 8 + jt) * 16;
      v8f c = {};
      for (int k0 = 0; k0 < 64; k0 += 4) {
        v2f a;
        a.x = kG[kvbase + (size_t)(m0 + l) * 64 + k0 + 2 * g];
        a.y = kG[kvbase + (size_t)(m0 + l) * 64 + k0 + 2 * g + 1];
        v2f bb;
        bb.x = qs[l * 64 + k0 + 2 * g];
        bb.y = qs[l * 64 + k0 + 2 * g + 1];
        c = __builtin_amdgcn_wmma_f32_16x16x4_f32(false, a, false, bb, (short)0, c, false, false);
      }
      if (l < 8) {
#pragma unroll
        for (int i = 0; i < 8; ++i) {
          const int n = m0 + i + 8 * g;
          const float d2 = kk[n] + misc[8 + l] - 2.f * c[i];
          Cs[n * 8 + l] = sqrtf(fmaxf(d2, 1e-12f));
        }
      }
    }
  }
  __syncthreads();

  // --- Phase D/E: MESH — 4 iterations of value_and_grad(entropy of Sinkhorn) ---
  for (int mi = 0; mi < 4; ++mi) {
    block_sink_fwd(Cs, la, lb, u, vv, uh, vh, red, tid, false);

    // P = exp(u + v - C); direct grads of H = -sum P log(P+eps)
    float gua[4];
    for (int j = 0; j < 4; ++j) {
      const int n = tid + j * 256;
      const float un = u[n];
      float acc = 0.f;
#pragma unroll
      for (int s = 0; s < 8; ++s) {
        const float p = __expf(un + vv[s] - Cs[n * 8 + s]);
        Pm[n * 8 + s] = p;
        const float gp = -(__logf(p + EPSF) + p / (p + EPSF));
        const float w = gp * p;             // dH/dK direct
        gC[n * 8 + s] = -w;                 // dH/dC = -dH/dK
        acc += w;
      }
      gua[j] = acc;                         // dH/du_8 direct
    }
    {
      float ps[8];
#pragma unroll
      for (int s = 0; s < 8; ++s) ps[s] = 0.f;
      for (int j = 0; j < 4; ++j) {
        const int n = tid + j * 256;
#pragma unroll
        for (int s = 0; s < 8; ++s) ps[s] += -gC[n * 8 + s];
      }
#pragma unroll
      for (int s = 0; s < 8; ++s) red[tid * 8 + s] = ps[s];
      __syncthreads();
      for (int st = 128; st > 0; st >>= 1) {
        if (tid < st)
#pragma unroll
          for (int s = 0; s < 8; ++s) red[tid * 8 + s] += red[(tid + st) * 8 + s];
        __syncthreads();
      }
      if (tid < 8) gvb[tid] = red[tid];     // dH/dv_8 direct
      __syncthreads();
    }
    // reverse through the 8 Sinkhorn iterations; softmaxes recomputed via
    //   S_t = exp(u_t + v_t - lb - C),  R_t = exp(u_t + v_{t-1} - la - C)
    for (int t = 8; t >= 1; --t) {
      float gvs[8];
#pragma unroll
      for (int s = 0; s < 8; ++s) gvs[s] = gvb[s];
      __syncthreads();
      for (int j = 0; j < 4; ++j) {       // v_t = f(K, u_t)
        const int n = tid + j * 256;
        const float ut = uh[(t - 1) * 1024 + n];
        float acc = 0.f;
#pragma unroll
        for (int s = 0; s < 8; ++s) {
          const float S = __expf(ut + vh[t * 8 + s] - lb[s] - Cs[n * 8 + s]);
          gC[n * 8 + s] += gvs[s] * S;
          acc += gvs[s] * S;
        }
        gua[j] -= acc;
      }
      float ps[8];
#pragma unroll
      for (int s = 0; s < 8; ++s) ps[s] = 0.f;
      for (int j = 0; j < 4; ++j) {       // u_t = g(K, v_{t-1})
        const int n = tid + j * 256;
        const float ut = uh[(t - 1) * 1024 + n];
        const float lan = la[n];
        const float gun = gua[j];
#pragma unroll
        for (int s = 0; s < 8; ++s) {
          const float R = __expf(ut + vh[(t - 1) * 8 + s] - lan - Cs[n * 8 + s]);
          gC[n * 8 + s] += gun * R;
          ps[s] += gun * R;
        }
        gua[j] = 0.f;
      }
#pragma unroll
      for (int s = 0; s < 8; ++s) red[tid * 8 + s] = ps[s];
      __syncthreads();
      for (int st = 128; st > 0; st >>= 1) {
        if (tid < st)
#pragma unroll
          for (int s = 0; s < 8; ++s) red[tid * 8 + s] += red[(tid + st) * 8 + s];
        __syncthreads();
      }
      if (tid < 8) gvb[tid] = -red[tid];
      __syncthreads();
    }
    // C <- C - lr * dH/dC
    for (int j = 0; j < 4; ++j) {
      const int n = tid + j * 256;
#pragma unroll
      for (int s = 0; s < 8; ++s) Cs[n * 8 + s] -= 3.0f * gC[n * 8 + s];
    }
    __syncthreads();
  }

  // --- Phase F: final Sinkhorn warm-started with (u, v) from last MESH fwd ---
  block_sink_fwd(Cs, la, lb, u, vv, nullptr, vh, red, tid, true);
  for (int j = 0; j < 4; ++j) {
    const int n = tid + j * 256;
    const float un = u[n];
#pragma unroll
    for (int s = 0; s < 8; ++s) {
      const float p = __expf(un + vv[s] - Cs[n * 8 + s]);
      Pm[n * 8 + s] = p;
      if (is_last) out_attn[(size_t)b * 8192 + s * 1024 + n] = p;
    }
  }
  __syncthreads();

  // --- Phase G: updates = attn @ v, with v staged into LDS through the
  // CDNA5 async global->LDS path (uh region reused as a 128x64 f32 tile). ---
  float* vtile = uh;  // 8192 floats free after MESH
  float accg[2] = {0.f, 0.f};
  for (int chunk = 0; chunk < 8; ++chunk) {
    const float* src = vG + kvbase + (size_t)chunk * 128 * 64;
    // 512 x 16B packets; each thread issues 2 async copies
#pragma unroll
    for (int p = 0; p < 2; ++p) {
      const int pkt = tid + p * 256;
      unsigned ldsoff = (unsigned)(unsigned long long)(vtile + pkt * 4);
      unsigned long long gaddr = (unsigned long long)(src + pkt * 4);
      asm volatile("global_load_async_to_lds_b128 %0, %1, off"
                   :: "v"(ldsoff), "v"(gaddr) : "memory");
    }
    if (chunk < 7) __builtin_prefetch(src + 8192, 0, 0);  // warm next tile in L2
    asm volatile("s_wait_asynccnt 0" ::: "memory");
    __syncthreads();
#pragma unroll
    for (int rep = 0; rep < 2; ++rep) {
      const int idx = tid + rep * 256;
      const int s = idx >> 6, d = idx & 63;
      float acc = accg[rep];
      for (int n = 0; n < 128; ++n)
        acc += Pm[(chunk * 128 + n) * 8 + s] * vtile[n * 64 + d];
      accg[rep] = acc;
    }
    __syncthreads();
  }
#pragma unroll
  for (int rep = 0; rep < 2; ++rep) {
    const int idx = tid + rep * 256;
    upd[idx] = accg[rep];
  }
  if (is_last && tid < 16) {
    const int s = tid >> 1, c = tid & 1;
    float acc = 0.f;
    for (int n = 0; n < 1024; ++n) {
      const float gc = (c == 0) ? (float)(n & 31) * (1.f / 31.f)
                                : (float)(n >> 5) * (1.f / 31.f);
      acc += Pm[n * 8 + s] * gc;
    }
    out_pos[b * 16 + s * 2 + c] = acc;
  }
  __syncthreads();

  // --- Phase H: GRU cell ---
  for (int rep = 0; rep < 2; ++rep) {
    const int idx = tid + rep * 256;
    const int s = idx >> 6, d = idx & 63;
    const float* x = upd + s * 64;
    const float* h = sprev + s * 64;
    float ir = b_ih[d], iz = b_ih[64 + d], inn = b_ih[128 + d];
    float hr = b_hh[d], hz = b_hh[64 + d], hn = b_hh[128 + d];
#pragma unroll
    for (int e = 0; e < 64; ++e) {
      const float xe = x[e], he = h[e];
      ir  += xe * w_ih[(size_t)d * 64 + e];
      iz  += xe * w_ih[(size_t)(64 + d) * 64 + e];
      inn += xe * w_ih[(size_t)(128 + d) * 64 + e];
      hr  += he * w_hh[(size_t)d * 64 + e];
      hz  += he * w_hh[(size_t)(64 + d) * 64 + e];
      hn  += he * w_hh[(size_t)(128 + d) * 64 + e];
    }
    const float r = 1.f / (1.f + __expf(-(ir + hr)));
    const float z = 1.f / (1.f + __expf(-(iz + hz)));
    const float nn = tanhf(inn + r * hn);
    hnew[s * 64 + d] = (1.f - z) * nn + z * h[d];
  }
  __syncthreads();

  // --- Phase I: residual MLP (LN -> fc1 -> relu -> fc2) ---
  if (tid < 8) {
    float m = 0.f;
#pragma unroll
    for (int i = 0; i < 64; ++i) m += hnew[tid * 64 + i];
    m *= (1.f / 64.f);
    float var = 0.f;
#pragma unroll
    for (int i = 0; i < 64; ++i) { const float d = hnew[tid * 64 + i] - m; var += d * d; }
    var *= (1.f / 64.f);
    misc[16 + tid] = m;
    misc[24 + tid] = rsqrtf(var + 1e-5f);
  }
  __syncthreads();
  float* ff = qs;  // reuse as [8][128]
  for (int j = 0; j < 4; ++j) {
    const int idx = tid + j * 256;
    const int s = idx >> 7, hh = idx & 127;
    const float m = misc[16 + s], rs = misc[24 + s];
    float acc = fc1_b[hh];
#pragma unroll
    for (int e = 0; e < 64; ++e) {
      const float y = (hnew[s * 64 + e] - m) * rs * ln_f_g[e] + ln_f_b[e];
      acc += y * fc1_w[(size_t)hh * 64 + e];
    }
    ff[idx] = fmaxf(acc, 0.f);
  }
  __syncthreads();
  for (int rep = 0; rep < 2; ++rep) {
    const int idx = tid + rep * 256;
    const int s = idx >> 6, d = idx & 63;
    float acc = fc2_b[d];
#pragma unroll
    for (int h = 0; h < 128; ++h) acc += ff[s * 128 + h] * fc2_w[(size_t)d * 128 + h];
    const float outv = hnew[s * 64 + d] + acc;
    slotsG[(size_t)b * 512 + idx] = outv;
    if (is_last) out_slots[(size_t)b * 512 + idx] = outv;
  }
}

// ---------------------------------------------------------------------------
extern "C" void kernel_launch(void* const* d_in, const int* in_sizes, int n_in,
                              void* d_out, int out_size, void* d_ws, size_t ws_size,
                              hipStream_t stream) {
  (void)in_sizes; (void)n_in; (void)out_size; (void)ws_size;
  const float* inp     = (const float*)d_in[0];
  const float* noise   = (const float*)d_in[1];
  const float* mu      = (const float*)d_in[2];
  const float* sigma   = (const float*)d_in[3];
  const float* Wq      = (const float*)d_in[4];
  const float* Wk      = (const float*)d_in[5];
  const float* Wv      = (const float*)d_in[6];
  const float* w_ih    = (const float*)d_in[7];
  const float* w_hh    = (const float*)d_in[8];
  const float* b_ih    = (const float*)d_in[9];
  const float* b_hh    = (const float*)d_in[10];
  const float* fc1_w   = (const float*)d_in[11];
  const float* fc1_b   = (const float*)d_in[12];
  const float* fc2_w   = (const float*)d_in[13];
  const float* fc2_b   = (const float*)d_in[14];
  const float* ln_in_g = (const float*)d_in[15];
  const float* ln_in_b = (const float*)d_in[16];
  const float* ln_s_g  = (const float*)d_in[17];
  const float* ln_s_b  = (const float*)d_in[18];
  const float* ln_f_g  = (const float*)d_in[19];
  const float* ln_f_b  = (const float*)d_in[20];
  const float* wi_w    = (const float*)d_in[21];
  const float* wi_b    = (const float*)d_in[22];
  const float* ws_w    = (const float*)d_in[23];
  const float* ws_b    = (const float*)d_in[24];

  float* ws     = (float*)d_ws;
  float* kbuf   = ws;                                   // 16.7M floats
  float* vbuf   = kbuf + (size_t)256 * 1024 * 64;       // 16.7M floats
  float* logits = vbuf + (size_t)256 * 1024 * 64;       // 262144
  float* log_a  = logits + (size_t)256 * 1024;          // 262144
  float* slots  = log_a + (size_t)256 * 1024;           // 131072

  float* out       = (float*)d_out;
  float* out_slots = out;                               // [256,8,64]
  float* out_pos   = out + 256 * 8 * 64;                // [256,8,2]
  float* out_attn  = out_pos + 256 * 8 * 2;             // [256,8,1024]

  k_ln_proj<<<2048, 256, 0, stream>>>(inp, Wk, Wv, wi_w, wi_b, ln_in_g, ln_in_b,
                                      kbuf, vbuf, logits);
  k_log_a<<<256, 256, 0, stream>>>(logits, log_a);
  k_slots_init<<<512, 256, 0, stream>>>(mu, sigma, noise, slots);

  const size_t smem = (size_t)41128 * sizeof(float);    // ~164.5 KB dynamic LDS
  hipFuncSetAttribute(reinterpret_cast<const void*>(k_slot_iter),
                      hipFuncAttributeMaxDynamicSharedMemorySize, (int)smem);
  for (int it = 0; it < 3; ++it) {
    k_slot_iter<<<256, 256, smem, stream>>>(
        kbuf, vbuf, log_a, slots, Wq, ws_w, ws_b, ln_s_g, ln_s_b,
        w_ih, w_hh, b_ih, b_hh, fc1_w, fc1_b, fc2_w, fc2_b,
        ln_f_g, ln_f_b, out_slots, out_pos, out_attn, it == 2 ? 1 : 0);
  }
}